// TokenChoiceTopKRouter_54219667145006
// MI455X (gfx1250) — compile-verified
//
#include <hip/hip_runtime.h>
#include <math.h>

// ---------------------------------------------------------------------------
// MoE token-choice top-8 router for MI455X (gfx1250, wave32).
//   logits = x[8192,4096] @ gate_weight[64,4096]^T   (fp32 V_WMMA_F32_16X16X4_F32)
//   scores = sigmoid(logits); top-8 on (scores+bias), gather unbiased scores,
//   normalize, *2.5; histogram of selected experts.
//
// Roofline: x traffic (134 MB) bounds runtime (~5.8 us @ 23.3 TB/s). The
// 4.3 GFLOP fp32 GEMM needs ~740 TFLOP/s to stay bandwidth-bound -> must run
// on the WMMA pipe (fp32 shape keeps exact numerics for the top-k selection).
// Weights (1 MB) are L2-resident and shared by all blocks.
//
// v2/v3: global->LDS staging uses GLOBAL_LOAD_ASYNC_TO_LDS_B128 (ASYNCcnt)
// instead of global->VGPR->LDS. Removes the 32 staging VGPRs that caused
// scratch spills in v1 and overlaps the DMA-style copy with WMMA compute.
// v3 fixes the builtin's parameter types: (global v4i*, lds v4i*, int, int).
// ---------------------------------------------------------------------------

#define N_TOKENS   8192
#define DIM        4096
#define NEXP       64
#define TOPK       8
#define ROUTE_SCL  2.5f

#define BLK_TOK    64                    // tokens per workgroup
#define NTHREADS   128                   // 4 wave32
#define KCHUNK     32                    // K tile staged in LDS
#define NCHUNK     (DIM / KCHUNK)        // 128
#define XS         36                    // padded LDS row stride (floats): 144B,
                                         // 16B aligned, conflict-free for 16 lanes

typedef __attribute__((ext_vector_type(2))) float v2f;
typedef __attribute__((ext_vector_type(8))) float v8f;
typedef __attribute__((ext_vector_type(4))) int   v4i;

// ---- async global->LDS b128 copy (no VGPR data path, ASYNCcnt-tracked) ----
__device__ __forceinline__ void async_copy_b128(const float* g, float* l) {
#if __has_builtin(__builtin_amdgcn_global_load_async_to_lds_b128)
    typedef __attribute__((address_space(1))) v4i* gp_t;   // global int4*
    typedef __attribute__((address_space(3))) v4i* lp_t;   // LDS int4*
    __builtin_amdgcn_global_load_async_to_lds_b128(
        (gp_t)(unsigned long long)g,
        (lp_t)(unsigned)(unsigned long long)l,   // LDS offset = addr[31:0]
        0, 0);
#else
    unsigned long long ga = (unsigned long long)g;
    unsigned           la = (unsigned)(unsigned long long)l;
    asm volatile("global_load_async_to_lds_b128 %0, %1, off"
                 :: "v"(la), "v"(ga) : "memory");
#endif
}

__device__ __forceinline__ void wait_async0() {
#if __has_builtin(__builtin_amdgcn_s_wait_asynccnt)
    __builtin_amdgcn_s_wait_asynccnt(0);
#else
    asm volatile("s_wait_asynccnt 0x0" ::: "memory");
#endif
}

__global__ void zero_hist_kernel(float* __restrict__ h) {
    h[threadIdx.x] = 0.0f;
}

__global__ __launch_bounds__(NTHREADS)
void router_topk_kernel(const float* __restrict__ x,
                        const float* __restrict__ gw,
                        const float* __restrict__ eb,
                        float* __restrict__ out_scores,   // [8192,8]
                        float* __restrict__ out_idx,      // [8192,8] (as float)
                        float* __restrict__ out_hist)     // [64]
{
    // GEMM stage: double-buffered x tile [2][64][36] + w tile [2][64][36]
    __shared__ __align__(16) unsigned char smem[2 * (BLK_TOK * XS + NEXP * XS) * 4];
    float* sX = (float*)smem;                 // [2][BLK_TOK*XS]
    float* sW = sX + 2 * BLK_TOK * XS;        // [2][NEXP*XS]

    const int tid  = threadIdx.x;
    const int lane = tid & 31;
    const int wv   = tid >> 5;       // wave id 0..3 -> 16-token slice
    const int half = lane >> 4;      // 0: K+0/K+1, 1: K+2/K+3 (A & B fp32 layout)
    const int lm   = lane & 15;      // M (A) / N (B) within tile
    const int tok0 = blockIdx.x * BLK_TOK;

    v8f acc[4] = {};                 // 16x16 fp32 C tiles for expert groups 0,16,32,48

    // ---- prologue: async-stage chunk 0 ------------------------------------
    {
        #pragma unroll
        for (int j = 0; j < 4; ++j) {
            const int idx = tid + j * NTHREADS;       // 0..511
            const int row = idx >> 3, c4 = idx & 7;
            async_copy_b128(x  + (size_t)(tok0 + row) * DIM + c4 * 4,
                            sX + row * XS + c4 * 4);
            async_copy_b128(gw + (size_t)row * DIM + c4 * 4,
                            sW + row * XS + c4 * 4);
        }
    }
    wait_async0();
    __syncthreads();

    // ---- main K loop: compute chunk c while async-fetching chunk c+1 ------
    for (int c = 0; c < NCHUNK; ++c) {
        const int  cur  = c & 1;
        const bool more = (c + 1) < NCHUNK;

        if (more) {
            const int nxt = cur ^ 1;
            const int kc  = (c + 1) * KCHUNK;
            float* dX = sX + nxt * (BLK_TOK * XS);
            float* dW = sW + nxt * (NEXP * XS);
            #pragma unroll
            for (int j = 0; j < 4; ++j) {
                const int idx = tid + j * NTHREADS;
                const int row = idx >> 3, c4 = idx & 7;
                async_copy_b128(x  + (size_t)(tok0 + row) * DIM + kc + c4 * 4,
                                dX + row * XS + c4 * 4);
                async_copy_b128(gw + (size_t)row * DIM + kc + c4 * 4,
                                dW + row * XS + c4 * 4);
            }
        }

        const float* bX = sX + cur * (BLK_TOK * XS);
        const float* bW = sW + cur * (NEXP * XS);
        #pragma unroll
        for (int k0 = 0; k0 < KCHUNK; k0 += 4) {
            // A fragment: token row = wv*16+lm, (K,K+1) lanes 0-15 / (K+2,K+3) lanes 16-31
            const float2 av = *(const float2*)(bX + (wv * 16 + lm) * XS + k0 + 2 * half);
            v2f a; a[0] = av.x; a[1] = av.y;
            #pragma unroll
            for (int t = 0; t < 4; ++t) {
                // B fragment: expert n = t*16+lm, same K split across lane halves
                const float2 bv = *(const float2*)(bW + (t * 16 + lm) * XS + k0 + 2 * half);
                v2f b; b[0] = bv.x; b[1] = bv.y;
                acc[t] = __builtin_amdgcn_wmma_f32_16x16x4_f32(
                    false, a, false, b, (short)0, acc[t], false, false);
            }
        }

        wait_async0();        // our async writes into the nxt buffer landed
        __syncthreads();      // everyone's landed; cur buffer free for reuse
    }

    // ---- epilogue: sigmoid -> LDS, top-8, normalize, histogram ------------
    // Overlay GEMM LDS (all waves past the final barrier).
    float* sS    = (float*)smem;            // [BLK_TOK][65] sigmoid scores
    float* sBias = sS + BLK_TOK * 65;       // [64]
    float* sCnt  = sBias + NEXP;            // [64] per-block histogram

    if (tid < NEXP) { sBias[tid] = eb[tid]; sCnt[tid] = 0.0f; }

    #pragma unroll
    for (int t = 0; t < 4; ++t) {
        #pragma unroll
        for (int v = 0; v < 8; ++v) {
            // C layout: lanes 0-15 -> M=v, lanes 16-31 -> M=v+8; N = lane&15
            const int m = wv * 16 + v + 8 * half;
            const int e = t * 16 + lm;
            const float sc = 1.0f / (1.0f + __expf(-acc[t][v]));
            sS[m * 65 + e] = sc;
        }
    }
    __syncthreads();

    if (tid < BLK_TOK) {
        const float* row = sS + tid * 65;
        unsigned long long chosen = 0ull;
        float ps[TOPK]; int pi[TOPK];
        float ssum = 0.0f;
        #pragma unroll
        for (int j = 0; j < TOPK; ++j) {
            float best = -3.0e38f; int bi = 0;
            for (int e = 0; e < NEXP; ++e) {
                if ((chosen >> e) & 1ull) continue;
                const float v = row[e] + sBias[e];   // bias picks, scores returned
                if (v > best) { best = v; bi = e; }  // strict > == lowest-index ties
            }
            chosen |= 1ull << bi;
            const float sc = row[bi];
            ps[j] = sc; pi[j] = bi; ssum += sc;
        }
        const float scl = ROUTE_SCL / (ssum + 1e-20f);
        const int g = tok0 + tid;
        #pragma unroll
        for (int j = 0; j < TOPK; ++j) {
            out_scores[g * TOPK + j] = ps[j] * scl;
            out_idx[g * TOPK + j]    = (float)pi[j];
            atomicAdd(&sCnt[pi[j]], 1.0f);           // ds_add_f32
        }
    }
    __syncthreads();
    if (tid < NEXP) atomicAdd(&out_hist[tid], sCnt[tid]);  // global_atomic_add_f32
}

extern "C" void kernel_launch(void* const* d_in, const int* in_sizes, int n_in,
                              void* d_out, int out_size, void* d_ws, size_t ws_size,
                              hipStream_t stream) {
    const float* x  = (const float*)d_in[0];   // [8192,4096]
    const float* gw = (const float*)d_in[1];   // [64,4096]
    const float* eb = (const float*)d_in[2];   // [64]

    float* out        = (float*)d_out;
    float* out_scores = out;                               // 8192*8
    float* out_idx    = out + (size_t)N_TOKENS * TOPK;     // 8192*8
    float* out_hist   = out + (size_t)N_TOKENS * TOPK * 2; // 64

    zero_hist_kernel<<<1, NEXP, 0, stream>>>(out_hist);
    router_topk_kernel<<<N_TOKENS / BLK_TOK, NTHREADS, 0, stream>>>(
        x, gw, eb, out_scores, out_idx, out_hist);
}